// Block_20109036879946
// MI455X (gfx1250) — compile-verified
//
#include <hip/hip_runtime.h>
#include <hip/hip_bf16.h>

#define D_MODEL 384
#define D_INNER 768
#define D_STATE 16
#define D_CONV  4
#define DT_RANK 24
#define SEQ_L   2048

typedef __attribute__((ext_vector_type(2))) float v2f;
typedef __attribute__((ext_vector_type(8))) float v8f;

enum { E_NONE = 0, E_BIAS_RELU = 1, E_BIAS_ADD = 2, E_ADD = 3, E_BIAS_SOFTPLUS = 4 };

// ---------------------------------------------------------------------------
// LayerNorm: one block per row, 128 threads, shfl + LDS reduction (wave32)
// ---------------------------------------------------------------------------
__global__ void ln_kernel(const float* __restrict__ x, const float* __restrict__ g,
                          const float* __restrict__ b, float* __restrict__ o, int D) {
    int row = blockIdx.x;
    const float* xr = x + (size_t)row * D;
    float s = 0.f, s2 = 0.f;
    for (int i = threadIdx.x; i < D; i += blockDim.x) {
        float v = xr[i];
        s += v; s2 += v * v;
    }
    for (int off = 16; off > 0; off >>= 1) {
        s  += __shfl_down(s,  off, 32);
        s2 += __shfl_down(s2, off, 32);
    }
    __shared__ float ss[4], sq[4];
    int wid = threadIdx.x >> 5, lane = threadIdx.x & 31;
    if (lane == 0) { ss[wid] = s; sq[wid] = s2; }
    __syncthreads();
    if (threadIdx.x == 0) {
        float ts = 0.f, tq = 0.f;
        for (int i = 0; i < 4; ++i) { ts += ss[i]; tq += sq[i]; }
        ss[0] = ts; sq[0] = tq;
    }
    __syncthreads();
    float inv = 1.f / (float)D;
    float mean = ss[0] * inv;
    float var  = sq[0] * inv - mean * mean;
    float rs = rsqrtf(var + 1e-5f);
    for (int i = threadIdx.x; i < D; i += blockDim.x) {
        float v = xr[i];
        o[(size_t)row * D + i] = (v - mean) * rs * g[i] + b[i];
    }
}

// ---------------------------------------------------------------------------
// FP32 WMMA GEMM:  C[M x N] = A[M x K] @ W[N x K]^T  (+ fused epilogue)
// One wave computes a 32x64 strip: 2 A-frags x 4 B-frags -> 8 accumulators.
// Branch-free inner loop: OOB N-columns clamp the W row pointer (in-bounds
// reads, garbage stays in its own column) and are skipped at store time.
// K loop is software-pipelined (prefetch k+4 before wmma of k) so the
// scheduler can issue partial s_wait_loadcnt and overlap VMEM with WMMA.
// Fragment layouts per CDNA5 ISA 7.12.2 (32-bit A 16x4 / B 4x16 / C 16x16).
// ---------------------------------------------------------------------------
template <int EPI>
__global__ __launch_bounds__(256)
void gemm_wmma(const float* __restrict__ A, int lda,
               const float* __restrict__ W,
               float* __restrict__ C, int ldc,
               int M, int N, int K,
               const float* __restrict__ bias,
               const float* __restrict__ res, int ldres) {
    const int Nt = (N + 63) >> 6;          // 64-wide N strips
    const int Mt = M >> 5;                 // 32-tall M strips
    int wave = (blockIdx.x * blockDim.x + threadIdx.x) >> 5;   // wave-uniform
    if (wave >= Mt * Nt) return;           // whole-wave exit: EXEC stays full
    int mt = wave / Nt, nt = wave % Nt;
    int m0 = mt << 5, n0 = nt << 6;
    int lane = threadIdx.x & 31;
    int half = lane >> 4;                  // K-half selector for A/B frags
    int r    = lane & 15;                  // M index (A) / N index (B, C/D)
    int koff = 2 * half;                   // lanes 0-15: K 0,1 ; 16-31: K 2,3

    v8f acc[8] = {v8f{}, v8f{}, v8f{}, v8f{}, v8f{}, v8f{}, v8f{}, v8f{}};

    const float* arow0 = A + (size_t)(m0 + r) * lda + koff;
    const float* arow1 = arow0 + (size_t)16 * lda;
    const float* wrow[4];
#pragma unroll
    for (int j = 0; j < 4; ++j) {
        int n = n0 + (j << 4) + r;
        int nc = n < N ? n : (N - 1);      // clamp: stay in-bounds, col unused
        wrow[j] = W + (size_t)nc * K + koff;
    }

    // prologue: fragments for k0 = 0
    v2f a0 = *(const v2f*)(arow0);
    v2f a1 = *(const v2f*)(arow1);
    v2f b[4];
#pragma unroll
    for (int j = 0; j < 4; ++j) b[j] = *(const v2f*)(wrow[j]);

    int k0 = 0;
    for (; k0 + 4 < K; k0 += 4) {
        // prefetch next k-step's fragments
        v2f a0n = *(const v2f*)(arow0 + k0 + 4);
        v2f a1n = *(const v2f*)(arow1 + k0 + 4);
        v2f bn[4];
#pragma unroll
        for (int j = 0; j < 4; ++j) bn[j] = *(const v2f*)(wrow[j] + k0 + 4);
        // compute current k-step
#pragma unroll
        for (int j = 0; j < 4; ++j) {
            acc[j]     = __builtin_amdgcn_wmma_f32_16x16x4_f32(
                false, a0, false, b[j], (short)0, acc[j],     false, false);
            acc[4 + j] = __builtin_amdgcn_wmma_f32_16x16x4_f32(
                false, a1, false, b[j], (short)0, acc[4 + j], false, false);
        }
        a0 = a0n; a1 = a1n;
#pragma unroll
        for (int j = 0; j < 4; ++j) b[j] = bn[j];
    }
    // epilogue k-step
#pragma unroll
    for (int j = 0; j < 4; ++j) {
        acc[j]     = __builtin_amdgcn_wmma_f32_16x16x4_f32(
            false, a0, false, b[j], (short)0, acc[j],     false, false);
        acc[4 + j] = __builtin_amdgcn_wmma_f32_16x16x4_f32(
            false, a1, false, b[j], (short)0, acc[4 + j], false, false);
    }

#pragma unroll
    for (int mi = 0; mi < 2; ++mi) {
#pragma unroll
        for (int j = 0; j < 4; ++j) {
            int n = n0 + (j << 4) + r;
            if (n >= N) continue;
#pragma unroll
            for (int v = 0; v < 8; ++v) {
                int m = m0 + mi * 16 + half * 8 + v;  // C/D: VGPR v -> row v+8*half
                float val = acc[mi * 4 + j][v];
                if (EPI == E_BIAS_RELU) {
                    val += bias[n];
                    val = fmaxf(val, 0.f);
                } else if (EPI == E_BIAS_ADD) {
                    val += bias[n] + res[(size_t)m * ldres + n];
                } else if (EPI == E_ADD) {
                    val += res[(size_t)m * ldres + n];
                } else if (EPI == E_BIAS_SOFTPLUS) {
                    val += bias[n];
                    val = fmaxf(val, 0.f) + log1pf(__expf(-fabsf(val)));
                }
                C[(size_t)m * ldc + n] = val;
            }
        }
    }
}

// ---------------------------------------------------------------------------
// Depthwise causal conv (width 4) + SiLU.  Reads xs = xz[:, :, 0:768].
// ---------------------------------------------------------------------------
__global__ void conv_silu_kernel(const float* __restrict__ xz, const float* __restrict__ cw,
                                 const float* __restrict__ cb, float* __restrict__ u,
                                 int Bn, int L) {
    int idx = blockIdx.x * blockDim.x + threadIdx.x;
    if (idx >= Bn * L * D_INNER) return;
    int d = idx % D_INNER;
    int t = (idx / D_INNER) % L;
    int b = idx / (D_INNER * L);
    float acc = cb[d];
    const float* base = xz + (size_t)b * L * (2 * D_INNER) + d;
#pragma unroll
    for (int j = 0; j < D_CONV; ++j) {
        int tt = t + j - (D_CONV - 1);
        if (tt >= 0) acc += base[(size_t)tt * (2 * D_INNER)] * cw[d * D_CONV + j];
    }
    float sig = 1.f / (1.f + __expf(-acc));
    u[idx] = acc * sig;   // idx == (b*L+t)*768 + d
}

// ---------------------------------------------------------------------------
// Selective scan: one thread per (b, d); h[16], A[16] in VGPRs.
// B/C rows fetched as float4 (b128) vectors; 1 v_exp per state per step
// (TRANS co-executes with VALU on the SIMD32s).
// Fused: y = (scan_y + u*D) * silu(res).
// ---------------------------------------------------------------------------
__global__ void scan_kernel(const float* __restrict__ dt, const float* __restrict__ u,
                            const float* __restrict__ dbc, const float* __restrict__ xz,
                            const float* __restrict__ A_log, const float* __restrict__ Dv,
                            float* __restrict__ y, int Bn, int L) {
    int idx = blockIdx.x * blockDim.x + threadIdx.x;
    if (idx >= Bn * D_INNER) return;
    int d = idx % D_INNER;
    int b = idx / D_INNER;

    float Amat[D_STATE], h[D_STATE];
#pragma unroll
    for (int n = 0; n < D_STATE; ++n) {
        Amat[n] = -__expf(A_log[d * D_STATE + n]);
        h[n] = 0.f;
    }
    float Dp = Dv[d];

    const int NDBC = DT_RANK + 2 * D_STATE;   // 56
    const float* dt_p  = dt  + (size_t)b * L * D_INNER + d;
    const float* u_p   = u   + (size_t)b * L * D_INNER + d;
    const float* dbc_p = dbc + (size_t)b * L * NDBC;
    const float* res_p = xz  + (size_t)b * L * (2 * D_INNER) + D_INNER + d;
    float*       y_p   = y   + (size_t)b * L * D_INNER + d;

    for (int t = 0; t < L; ++t) {
        float dtv = dt_p[(size_t)t * D_INNER];
        float uv  = u_p[(size_t)t * D_INNER];
        float rv  = res_p[(size_t)t * (2 * D_INNER)];
        float du  = dtv * uv;

        // B at row+24 (16B aligned: 24*4=96, row stride 56*4=224), C at row+40
        const float4* bcp = (const float4*)(dbc_p + (size_t)t * NDBC + DT_RANK);
        float Bv[D_STATE], Cv[D_STATE];
        ((float4*)Bv)[0] = bcp[0]; ((float4*)Bv)[1] = bcp[1];
        ((float4*)Bv)[2] = bcp[2]; ((float4*)Bv)[3] = bcp[3];
        ((float4*)Cv)[0] = bcp[4]; ((float4*)Cv)[1] = bcp[5];
        ((float4*)Cv)[2] = bcp[6]; ((float4*)Cv)[3] = bcp[7];

        float acc = 0.f;
#pragma unroll
        for (int n = 0; n < D_STATE; ++n) {
            float dA = __expf(dtv * Amat[n]);
            h[n] = dA * h[n] + du * Bv[n];
            acc += h[n] * Cv[n];
        }
        acc += uv * Dp;
        float sig = 1.f / (1.f + __expf(-rv));
        y_p[(size_t)t * D_INNER] = acc * (rv * sig);
    }
}

// ---------------------------------------------------------------------------
// Host-side dispatch
// ---------------------------------------------------------------------------
static void launch_gemm(int epi, const float* A, int lda, const float* W,
                        float* C, int ldc, int M, int N, int K,
                        const float* bias, const float* res, int ldres,
                        hipStream_t s) {
    int waves = (M >> 5) * ((N + 63) >> 6);
    int blocks = (waves + 7) / 8;                // 8 waves (256 threads) / block
    switch (epi) {
    case E_NONE:
        gemm_wmma<E_NONE><<<blocks, 256, 0, s>>>(A, lda, W, C, ldc, M, N, K, bias, res, ldres); break;
    case E_BIAS_RELU:
        gemm_wmma<E_BIAS_RELU><<<blocks, 256, 0, s>>>(A, lda, W, C, ldc, M, N, K, bias, res, ldres); break;
    case E_BIAS_ADD:
        gemm_wmma<E_BIAS_ADD><<<blocks, 256, 0, s>>>(A, lda, W, C, ldc, M, N, K, bias, res, ldres); break;
    case E_ADD:
        gemm_wmma<E_ADD><<<blocks, 256, 0, s>>>(A, lda, W, C, ldc, M, N, K, bias, res, ldres); break;
    case E_BIAS_SOFTPLUS:
        gemm_wmma<E_BIAS_SOFTPLUS><<<blocks, 256, 0, s>>>(A, lda, W, C, ldc, M, N, K, bias, res, ldres); break;
    }
}

extern "C" void kernel_launch(void* const* d_in, const int* in_sizes, int n_in,
                              void* d_out, int out_size, void* d_ws, size_t ws_size,
                              hipStream_t stream) {
    const float* x        = (const float*)d_in[0];
    const float* ln1_g    = (const float*)d_in[1];
    const float* ln1_b    = (const float*)d_in[2];
    const float* ln2_g    = (const float*)d_in[3];
    const float* ln2_b    = (const float*)d_in[4];
    const float* in_proj  = (const float*)d_in[5];
    const float* conv_w   = (const float*)d_in[6];
    const float* conv_b   = (const float*)d_in[7];
    const float* x_proj   = (const float*)d_in[8];
    const float* dtp_w    = (const float*)d_in[9];
    const float* dtp_b    = (const float*)d_in[10];
    const float* A_log    = (const float*)d_in[11];
    const float* Dvec     = (const float*)d_in[12];
    const float* out_proj = (const float*)d_in[13];
    const float* ffn_w1   = (const float*)d_in[14];
    const float* ffn_b1   = (const float*)d_in[15];
    const float* ffn_w2   = (const float*)d_in[16];
    const float* ffn_b2   = (const float*)d_in[17];
    float* out = (float*)d_out;

    const int rows = in_sizes[0] / D_MODEL;   // B*L = 8192
    const int L = SEQ_L;
    const int Bn = rows / L;                  // 4
    const int NDBC = DT_RANK + 2 * D_STATE;   // 56

    // workspace carve-up (fp32)
    float* ws   = (float*)d_ws;
    float* xln  = ws;                                  // rows*384
    float* xz   = xln + (size_t)rows * D_MODEL;        // rows*1536
    float* u    = xz  + (size_t)rows * 2 * D_INNER;    // rows*768
    float* dbc  = u   + (size_t)rows * D_INNER;        // rows*56
    float* dt   = dbc + (size_t)rows * NDBC;           // rows*768
    float* yb   = dt  + (size_t)rows * D_INNER;        // rows*768
    float* xres = yb  + (size_t)rows * D_INNER;        // rows*384
    float* hbuf = xz;                                  // FFN hidden reuses xz

    // 1) LN1
    ln_kernel<<<rows, 128, 0, stream>>>(x, ln1_g, ln1_b, xln, D_MODEL);
    // 2) in_proj: xz = xln @ in_proj^T           [rows x 1536]
    launch_gemm(E_NONE, xln, D_MODEL, in_proj, xz, 2 * D_INNER,
                rows, 2 * D_INNER, D_MODEL, nullptr, nullptr, 0, stream);
    // 3) depthwise causal conv + SiLU -> u       [rows x 768]
    {
        int tot = Bn * L * D_INNER;
        conv_silu_kernel<<<(tot + 255) / 256, 256, 0, stream>>>(xz, conv_w, conv_b, u, Bn, L);
    }
    // 4) x_proj: dbc = u @ x_proj^T              [rows x 56] (N clamped/guarded)
    launch_gemm(E_NONE, u, D_INNER, x_proj, dbc, NDBC,
                rows, NDBC, D_INNER, nullptr, nullptr, 0, stream);
    // 5) dt = softplus(dbc[:, :24] @ dtp_w^T + dtp_b)   [rows x 768]
    launch_gemm(E_BIAS_SOFTPLUS, dbc, NDBC, dtp_w, dt, D_INNER,
                rows, D_INNER, DT_RANK, dtp_b, nullptr, 0, stream);
    // 6) selective scan (fused +u*D and *silu(res)) -> yb
    {
        int tot = Bn * D_INNER;
        scan_kernel<<<(tot + 255) / 256, 256, 0, stream>>>(dt, u, dbc, xz, A_log, Dvec, yb, Bn, L);
    }
    // 7) out_proj + residual: xres = x + yb @ out_proj^T
    launch_gemm(E_ADD, yb, D_INNER, out_proj, xres, D_MODEL,
                rows, D_MODEL, D_INNER, nullptr, x, D_MODEL, stream);
    // 8) LN2 -> xln (reuse)
    ln_kernel<<<rows, 128, 0, stream>>>(xres, ln2_g, ln2_b, xln, D_MODEL);
    // 9) FFN1: hbuf = relu(xln @ ffn_w1^T + b1)  [rows x 1536]
    launch_gemm(E_BIAS_RELU, xln, D_MODEL, ffn_w1, hbuf, 4 * D_MODEL,
                rows, 4 * D_MODEL, D_MODEL, ffn_b1, nullptr, 0, stream);
    // 10) FFN2 + bias + residual -> out
    launch_gemm(E_BIAS_ADD, hbuf, 4 * D_MODEL, ffn_w2, out, D_MODEL,
                rows, D_MODEL, 4 * D_MODEL, ffn_b2, xres, D_MODEL, stream);
}